// SE3CompositeLayer_31439160607076
// MI455X (gfx1250) — compile-verified
//
#include <hip/hip_runtime.h>
#include <hip/hip_bf16.h>
#include <math.h>

// SE3 composite: per-row Rodrigues -> quaternion -> Euler. Memory-bandwidth
// bound (~100MB total @ 23.3 TB/s ~ 4.3us). Streamed via CDNA5 async
// global<->LDS b128 copies (ASYNCcnt-tracked), wave-local so no barriers.

#define THREADS 256
#define ROWS_PER_THREAD 2
#define ROWS_PER_BLOCK (THREADS * ROWS_PER_THREAD)      // 512 rows
#define BYTES_PER_ROW 24                                 // 6 f32
#define TILE_BYTES (ROWS_PER_BLOCK * BYTES_PER_ROW)      // 12288 B
#define WAVE_BYTES (64 * BYTES_PER_ROW)                  // 1536 B per wave (64 rows)

__device__ __forceinline__ void se3_row(const float* __restrict__ v,
                                        float* __restrict__ o) {
    const float ux = v[0], uy = v[1], uz = v[2];
    const float wx = v[3], wy = v[4], wz = v[5];

    const float theta2 = wx * wx + wy * wy + wz * wz;
    const float theta  = sqrtf(theta2);
    const float A  = sinf(theta) / theta;
    const float Bc = (1.0f - cosf(theta)) / (theta * theta);

    const float xx = wx * wx, yy = wy * wy, zz = wz * wz;
    const float xy = wx * wy, xz = wx * wz, yz = wy * wz;

    // R = A*K + Bc*K2 + I, with K2 entries as the reference einsum produces.
    const float R00 = 1.0f + Bc * (-zz - yy);
    const float R01 = -A * wz + Bc * xy;
    const float R02 =  A * wy + Bc * xz;
    const float R10 =  A * wz + Bc * xy;
    const float R11 = 1.0f + Bc * (-zz - xx);
    const float R12 = -A * wx + Bc * yz;
    const float R20 = -A * wy + Bc * xz;
    const float R21 =  A * wx + Bc * yz;
    const float R22 = 1.0f + Bc * (-yy - xx);

    const float tr = R00 + R11 + R22;
    const float qw = sqrtf(tr + 1.0f) * 0.5f;
    const float qx = (R21 - R12) / (4.0f * qw);
    const float qy = (R02 - R20) / (4.0f * qw);
    const float qz = (R10 - R01) / (4.0f * qw);

    const float ysqr = qy * qy;
    const float roll = atan2f(2.0f * (qw * qx + qy * qz),
                              1.0f - 2.0f * (qx * qx + ysqr));
    float t2 = 2.0f * (qw * qy - qz * qx);
    t2 = fminf(fmaxf(t2, -1.0f), 1.0f);
    const float pitch = asinf(t2);
    const float yaw = atan2f(2.0f * (qw * qz + qx * qy),
                             1.0f - 2.0f * (ysqr + qz * qz));

    o[0] = roll; o[1] = pitch; o[2] = yaw;
    o[3] = ux;   o[4] = uy;    o[5] = uz;
}

__global__ __launch_bounds__(THREADS)
void se3_stream_kernel(const float* __restrict__ in,
                       float* __restrict__ out,
                       int nrows) {
    __shared__ float lin [ROWS_PER_BLOCK * 6];   // 12 KB staging in
    __shared__ float lout[ROWS_PER_BLOCK * 6];   // 12 KB staging out

    const int t    = threadIdx.x;
    const int wave = t >> 5;
    const int lane = t & 31;

    const long long tileRow0   = (long long)blockIdx.x * ROWS_PER_BLOCK;
    const int       rowsInTile = (int)min((long long)ROWS_PER_BLOCK,
                                          (long long)nrows - tileRow0);

    if (rowsInTile == ROWS_PER_BLOCK) {
        // ---- fast path: full tile, async-staged through LDS ----
        const size_t   tileByte = (size_t)blockIdx.x * TILE_BYTES;
        const unsigned waveByte = (unsigned)wave * WAVE_BYTES;
        const unsigned linBase  = (unsigned)(uintptr_t)(&lin[0]);
        const unsigned loutBase = (unsigned)(uintptr_t)(&lout[0]);

        // Async load: 3 x b128 per lane; each instruction is a contiguous
        // 512B wave transaction. Lanes read back rows other lanes of the wave loaded,
        // all within this wave's 1536B slice -> wave-local sync only.
        const char* gin = (const char*)in + tileByte + waveByte;
#pragma unroll
        for (int k = 0; k < 3; ++k) {
            const unsigned chunk = 16u * (unsigned)(lane + 32 * k);
            const unsigned lofs  = linBase + waveByte + chunk;
            const char*    g     = gin + chunk;
            asm volatile("global_load_async_to_lds_b128 %0, %1, off"
                         :: "v"(lofs), "v"(g) : "memory");
        }
        asm volatile("s_wait_asynccnt 0" ::: "memory");

        // Compute this thread's 2 rows straight out of LDS.
        float res[12];
        const float* src = &lin[t * 12];
        se3_row(src,     res);
        se3_row(src + 6, res + 6);

        float* dst = &lout[t * 12];
#pragma unroll
        for (int i = 0; i < 12; ++i) dst[i] = res[i];

        // Make DS stores visible to the async engine, then stream out.
        asm volatile("s_wait_dscnt 0" ::: "memory");
        char* gout = (char*)out + tileByte + waveByte;
#pragma unroll
        for (int k = 0; k < 3; ++k) {
            const unsigned chunk = 16u * (unsigned)(lane + 32 * k);
            const unsigned lofs  = loutBase + waveByte + chunk;
            char*          g     = gout + chunk;
            asm volatile("global_store_async_from_lds_b128 %0, %1, off"
                         :: "v"(g), "v"(lofs) : "memory");
        }
        asm volatile("s_wait_asynccnt 0" ::: "memory");
    } else {
        // ---- tail path (unused when nrows % 512 == 0) ----
        for (long long r = tileRow0 + t; r < (long long)nrows; r += THREADS) {
            float v[6], o[6];
            const float* p = in + r * 6;
#pragma unroll
            for (int i = 0; i < 6; ++i) v[i] = p[i];
            se3_row(v, o);
            float* q = out + r * 6;
#pragma unroll
            for (int i = 0; i < 6; ++i) q[i] = o[i];
        }
    }
}

extern "C" void kernel_launch(void* const* d_in, const int* in_sizes, int n_in,
                              void* d_out, int out_size, void* d_ws, size_t ws_size,
                              hipStream_t stream) {
    (void)n_in; (void)d_ws; (void)ws_size; (void)out_size;
    const float* in  = (const float*)d_in[0];
    float*       out = (float*)d_out;
    const int nrows  = in_sizes[0] / 6;                 // (B, 6) rows
    const int blocks = (nrows + ROWS_PER_BLOCK - 1) / ROWS_PER_BLOCK;
    se3_stream_kernel<<<blocks, THREADS, 0, stream>>>(in, out, nrows);
}